// NavGCM_25812753449242
// MI455X (gfx1250) — compile-verified
//
#include <hip/hip_runtime.h>
#include <stdint.h>

typedef __attribute__((ext_vector_type(2))) float v2f;
typedef __attribute__((ext_vector_type(8))) float v8f;

#define BATCH 2048
#define TAU   32
#define T0C   96
#define LROW  128
#define DD    64
#define PDIM  3
#define HD    68      // DD + PDIM + 1
#define DOUT  64
#define KNBR  16
#define HS    68      // LDS row stride for h (floats); 68*4=272B -> 16B aligned rows

// flat output offsets (elements): padded_output, new_x, new_pos, new_rot
#define OUT_PAD 0
#define OUT_X   (BATCH*TAU*DOUT)
#define OUT_POS (OUT_X + BATCH*LROW*DD)
#define OUT_ROT (OUT_POS + BATCH*LROW*PDIM)

__global__ __launch_bounds__(256) void navgcm_kernel(
    const float* __restrict__ x,      const float* __restrict__ pos,
    const float* __restrict__ rot,    const float* __restrict__ old_x,
    const float* __restrict__ old_pos,const float* __restrict__ old_rot,
    const float* __restrict__ Wself,  const float* __restrict__ Wnbr,
    const float* __restrict__ bias,   float* __restrict__ out)
{
  __shared__ float hbuf[LROW * HS];        // h = [x | pos | rot], later reused as msg (128x64)
  __shared__ int   idxbuf[TAU * KNBR];
  __shared__ int   cntbuf[TAU];

  const int b    = blockIdx.x;
  const int t    = threadIdx.x;
  const int lane = t & 31;
  const int w    = t >> 5;
  const int ln16 = lane & 15;
  const int hi   = lane >> 4;

  // ---------- Phase 0: scatter-update, emit new_x/new_pos/new_rot, stage h in LDS ----------
  {
    const float4* xv = (const float4*)(x     + (size_t)b * TAU  * DD);
    const float4* ox = (const float4*)(old_x + (size_t)b * LROW * DD);
    float4*       nx = (float4*)(out + OUT_X + (size_t)b * LROW * DD);
    for (int v = t; v < LROW * DD / 4; v += 256) {
      int r = v >> 4, c4 = v & 15;
      float4 val = (r >= T0C) ? xv[(r - T0C) * 16 + c4] : ox[v];
      nx[v] = val;
      *(float4*)&hbuf[r * HS + c4 * 4] = val;
    }
    for (int v = t; v < LROW * PDIM; v += 256) {
      int r = v / 3, c = v - r * 3;
      float val = (r >= T0C) ? pos[(size_t)b * TAU * PDIM + (r - T0C) * 3 + c]
                             : old_pos[(size_t)b * LROW * PDIM + v];
      out[OUT_POS + (size_t)b * LROW * PDIM + v] = val;
      hbuf[r * HS + DD + c] = val;
    }
    if (t < LROW) {
      float val = (t >= T0C) ? rot[(size_t)b * TAU + (t - T0C)]
                             : old_rot[(size_t)b * LROW + t];
      out[OUT_ROT + (size_t)b * LROW + t] = val;
      hbuf[t * HS + DD + PDIM] = val;
    }
  }
  __syncthreads();

  // ---------- Phase 1: causal radius top-K for query rows 96..127 (4 rows per wave) ----------
  {
    const uint32_t INFB = 0x7F800000u;
    for (int q = 0; q < 4; ++q) {
      int i  = T0C + w * 4 + q;
      int rq = i - T0C;
      float px = hbuf[i * HS + DD + 0];
      float py = hbuf[i * HS + DD + 1];
      float pz = hbuf[i * HS + DD + 2];
      float d[4];
      #pragma unroll
      for (int s = 0; s < 4; ++s) {
        int j = lane + 32 * s;
        float d2 = __builtin_inff();
        if (j < i) {
          float dx = px - hbuf[j * HS + DD + 0];
          float dy = py - hbuf[j * HS + DD + 1];
          float dz = pz - hbuf[j * HS + DD + 2];
          float v = dx * dx + dy * dy + dz * dz;
          if (v <= 1.0f) d2 = v;
        }
        d[s] = d2;
      }
      int cnt = 0;
      for (int k = 0; k < KNBR; ++k) {
        float best = __builtin_inff(); int bs = 0;
        #pragma unroll
        for (int s = 0; s < 4; ++s) { if (d[s] < best) { best = d[s]; bs = s; } }
        unsigned long long key =
            ((unsigned long long)__float_as_uint(best) << 32) | (unsigned)(lane + 32 * bs);
        #pragma unroll
        for (int off = 16; off > 0; off >>= 1) {
          unsigned long long o = __shfl_xor(key, off, 32);
          if (o < key) key = o;
        }
        if ((uint32_t)(key >> 32) >= INFB) break;   // uniform across wave
        int jmin = (int)(uint32_t)key;
        if ((jmin & 31) == lane) d[jmin >> 5] = __builtin_inff();
        if (lane == 0) idxbuf[rq * KNBR + cnt] = jmin;
        ++cnt;
      }
      if (lane == 0) cntbuf[rq] = cnt;
    }
  }
  __syncthreads();

  // ---------- Phase 2: f32 WMMA GEMMs, fully register-accumulated ----------
  // wave w: msg M-tile M0=16w (all 4 N-tiles) + one 16x16 self tile (rows mS.., cols nS..)
  v8f accM0 = {}, accM1 = {}, accM2 = {}, accM3 = {}, accS = {};
  const int M0 = 16 * w;
  const int mS = T0C + 16 * (w >> 2);
  const int nS = 16 * (w & 3);

  for (int k0 = 0; k0 < HD; k0 += 4) {
    int kA = k0 + 2 * hi;                       // 16x4 f32 A layout: VGPR v, half 'hi' -> K=k0+2*hi+v
    const float* hr = &hbuf[(M0 + ln16) * HS + kA];
    v2f A;  A.x  = hr[0];  A.y  = hr[1];
    const float* hs = &hbuf[(mS + ln16) * HS + kA];
    v2f As; As.x = hs[0];  As.y = hs[1];

    v2f Bs; Bs.x = Wself[kA * DOUT + nS + ln16]; Bs.y = Wself[(kA + 1) * DOUT + nS + ln16];
    accS = __builtin_amdgcn_wmma_f32_16x16x4_f32(false, As, false, Bs, (short)0, accS, false, false);

    v2f B0; B0.x = Wnbr[kA * DOUT +  0 + ln16]; B0.y = Wnbr[(kA + 1) * DOUT +  0 + ln16];
    accM0 = __builtin_amdgcn_wmma_f32_16x16x4_f32(false, A, false, B0, (short)0, accM0, false, false);
    v2f B1; B1.x = Wnbr[kA * DOUT + 16 + ln16]; B1.y = Wnbr[(kA + 1) * DOUT + 16 + ln16];
    accM1 = __builtin_amdgcn_wmma_f32_16x16x4_f32(false, A, false, B1, (short)0, accM1, false, false);
    v2f B2; B2.x = Wnbr[kA * DOUT + 32 + ln16]; B2.y = Wnbr[(kA + 1) * DOUT + 32 + ln16];
    accM2 = __builtin_amdgcn_wmma_f32_16x16x4_f32(false, A, false, B2, (short)0, accM2, false, false);
    v2f B3; B3.x = Wnbr[kA * DOUT + 48 + ln16]; B3.y = Wnbr[(kA + 1) * DOUT + 48 + ln16];
    accM3 = __builtin_amdgcn_wmma_f32_16x16x4_f32(false, A, false, B3, (short)0, accM3, false, false);
  }
  __syncthreads();   // everyone done reading h -> safe to overwrite with msg

  float* msgb = hbuf;                          // reuse LDS: msg is 128x64 (stride 64)
  #pragma unroll
  for (int g = 0; g < 8; ++g) {
    int r = M0 + g + 8 * hi;                   // C/D layout: VGPR g -> row M0+g (lanes 0-15), M0+8+g (16-31)
    msgb[r * DOUT +  0 + ln16] = accM0[g];
    msgb[r * DOUT + 16 + ln16] = accM1[g];
    msgb[r * DOUT + 32 + ln16] = accM2[g];
    msgb[r * DOUT + 48 + ln16] = accM3[g];
  }
  __syncthreads();

  // ---------- Phase 3: neighbor gather + bias + ReLU -> padded_output ----------
  #pragma unroll
  for (int g = 0; g < 8; ++g) {
    int r  = mS + g + 8 * hi;
    int rq = r - T0C;
    int n  = nS + ln16;
    float agg = 0.0f;
    int cnt = cntbuf[rq];
    for (int k = 0; k < cnt; ++k)
      agg += msgb[idxbuf[rq * KNBR + k] * DOUT + n];
    float val = accS[g] + agg + bias[n];
    out[OUT_PAD + (size_t)b * TAU * DOUT + rq * DOUT + n] = fmaxf(val, 0.0f);
  }
}

extern "C" void kernel_launch(void* const* d_in, const int* in_sizes, int n_in,
                              void* d_out, int out_size, void* d_ws, size_t ws_size,
                              hipStream_t stream) {
  (void)in_sizes; (void)n_in; (void)d_ws; (void)ws_size; (void)out_size;
  const float* x       = (const float*)d_in[0];
  const float* pos     = (const float*)d_in[1];
  const float* rot     = (const float*)d_in[2];
  const float* old_x   = (const float*)d_in[3];
  const float* old_pos = (const float*)d_in[4];
  const float* old_rot = (const float*)d_in[5];
  const float* Wself   = (const float*)d_in[6];
  const float* Wnbr    = (const float*)d_in[7];
  const float* bias    = (const float*)d_in[8];
  float* out = (float*)d_out;

  navgcm_kernel<<<dim3(BATCH), dim3(256), 0, stream>>>(
      x, pos, rot, old_x, old_pos, old_rot, Wself, Wnbr, bias, out);
}